// HNetReference_38422777430603
// MI455X (gfx1250) — compile-verified
//
#include <hip/hip_runtime.h>
#include <hip/hip_bf16.h>

typedef __attribute__((ext_vector_type(2))) float v2f;
typedef __attribute__((ext_vector_type(8))) float v8f;
typedef __attribute__((ext_vector_type(4))) unsigned int u32x4;
typedef __attribute__((ext_vector_type(4))) int i32x4;
typedef __attribute__((ext_vector_type(8))) int i32x8;

#define BSZ     2
#define LSEQ    4096
#define DMOD    1024
#define CHLEN   16                 // chunk length = WMMA tile M
#define NCHMAX  (LSEQ / CHLEN)     // 256 chunks max per batch
#define TGROUPS 8                  // channel tile-groups per (b,c)
#define TGW     128                // channels per tile-group (8 waves x 16)
#define HPAD    132                // H row stride in floats (bank-conflict-free)
#define WPAD    17                 // W row stride in floats (bank-conflict-free)

// Tensor Data Mover path (gather-mode row DMA into LDS), guarded so compile
// can never regress if the builtin is absent on this toolchain.
#if defined(__HIP_DEVICE_COMPILE__) && \
    __has_builtin(__builtin_amdgcn_tensor_load_to_lds) && \
    __has_builtin(__builtin_amdgcn_s_wait_tensorcnt)
#define USE_TDM 1
#else
#define USE_TDM 0
#endif

// ---- workspace layout (bytes) ----
static constexpr size_t SELIDX_OFF = 0;                         // int  [2][4096]
static constexpr size_t PCOEF_OFF  = SELIDX_OFF + 32768;        // f32  [2][4096]
static constexpr size_t DECROW_OFF = PCOEF_OFF  + 32768;        // f32  [2][4096]
static constexpr size_t IDXMAP_OFF = DECROW_OFF + 32768;        // int  [2][4096]
static constexpr size_t KARR_OFF   = IDXMAP_OFF + 32768;        // int  [2] (+pad)
static constexpr size_t ACHUNK_OFF = KARR_OFF   + 256;          // f32  [2][256]
static constexpr size_t CSTATE_OFF = ACHUNK_OFF + 2048;         // f32  [2][256][1024]
static constexpr size_t CARRY_OFF  = CSTATE_OFF + 2097152;      // f32  [2][256][1024]
static constexpr size_t YBUF_OFF   = CARRY_OFF  + 2097152;      // f32  [2][4096][1024]
// total ~36.1 MiB

// ---------------------------------------------------------------------------
// Stage 1: per-batch prefix scan over boundary_mask; emit compressed coeffs.
// ---------------------------------------------------------------------------
__global__ void __launch_bounds__(1024)
k_setup(const float* __restrict__ prob, const unsigned char* __restrict__ mask,
        int* __restrict__ selIdx, float* __restrict__ pcoef,
        int* __restrict__ idxmap, int* __restrict__ Karr) {
  const int b   = blockIdx.x;
  const int tid = threadIdx.x;
  __shared__ int scan[1024];

  const unsigned char* m  = mask + (size_t)b * LSEQ;
  const float*         pr = prob + (size_t)b * LSEQ;

  const int t0 = tid * 4;
  int mv[4], loc[4], tot = 0;
#pragma unroll
  for (int i = 0; i < 4; i++) { mv[i] = m[t0 + i] ? 1 : 0; tot += mv[i]; loc[i] = tot; }

  scan[tid] = tot;
  __syncthreads();
  for (int off = 1; off < 1024; off <<= 1) {        // Hillis-Steele inclusive
    int v = (tid >= off) ? scan[tid - off] : 0;
    __syncthreads();
    scan[tid] += v;
    __syncthreads();
  }
  const int excl = scan[tid] - tot;

#pragma unroll
  for (int i = 0; i < 4; i++) {
    const int t   = t0 + i;
    const int cnt = excl + loc[i];                  // inclusive cumsum of mask
    int j = cnt - 1; if (j < 0) j = 0;
    idxmap[(size_t)b * LSEQ + t] = j;               // clip(cumsum-1, 0, L-1)
    if (mv[i]) {
      float p = pr[t];
      p = fminf(fmaxf(p, 1e-4f), 1.0f - 1e-4f);
      selIdx[(size_t)b * LSEQ + (cnt - 1)] = t;
      pcoef [(size_t)b * LSEQ + (cnt - 1)] = p;
    }
  }
  if (tid == 1023) Karr[b] = scan[1023];
}

// ---------------------------------------------------------------------------
// Stage 2: per-chunk matrix-form scan via V_WMMA_F32_16X16X4_F32.
//   Y_local(16 x 128) = W(16x16, lower-tri) x H(16x128), chained K=4 WMMAs.
//   H tile gathered into LDS by the Tensor Data Mover (gather mode, 16 row
//   indices) when available; DMA overlapped with W-matrix construction.
// ---------------------------------------------------------------------------
__global__ void __launch_bounds__(256)
k_chunk(const float* __restrict__ hid, const int* __restrict__ selIdx,
        const float* __restrict__ pcoef, const int* __restrict__ Karr,
        float* __restrict__ Ybuf, float* __restrict__ cstate,
        float* __restrict__ Achunk, float* __restrict__ decrow) {
  const int tg = blockIdx.x, c = blockIdx.y, b = blockIdx.z;
  const int K = Karr[b];
  const int nchunk = (K + CHLEN - 1) / CHLEN;
  if (c >= nchunk) return;

  __shared__ float aS[CHLEN], pS[CHLEN];
  __shared__ int   tokS[CHLEN];
  __shared__ float W[CHLEN][WPAD];
  __shared__ __align__(128) float H[CHLEN][HPAD];

  const int tid = threadIdx.x;

  if (tid < CHLEN) {                                // chunk coefficients (pad: p=0,a=1)
    const int j = c * CHLEN + tid;
    float p = 0.0f; int tok = 0;
    if (j < K) { p = pcoef[(size_t)b * LSEQ + j]; tok = selIdx[(size_t)b * LSEQ + j]; }
    pS[tid] = p; aS[tid] = 1.0f - p; tokS[tid] = tok;
  }
  __syncthreads();

#if USE_TDM
  // Wave 0 issues one gather-mode TDM op: 16 rows (runtime indices) of 128
  // f32 each, global -> LDS, with LDS padding 4 dwords per 128-dword row so
  // the landed layout matches H[16][HPAD]. Tracked by TENSORcnt.
  if (tid < 32) {
    int tok[16];
#pragma unroll
    for (int k = 0; k < 16; k++) {
      const int j = c * CHLEN + k;
      tok[k] = (j < K) ? selIdx[(size_t)b * LSEQ + j] : 0;
    }
    const unsigned long long ga =
        (unsigned long long)(const void*)(hid + (size_t)b * LSEQ * DMOD + (size_t)tg * TGW);
    const unsigned int ldsoff = (unsigned int)(size_t)(&H[0][0]);
    u32x4 g0;
    g0[0] = 0x80000001u;                            // count=1, gather_mode, 16-bit idx
    g0[1] = ldsoff;                                 // lds_addr
    g0[2] = (unsigned int)ga;                       // global_addr[31:0]
    g0[3] = (unsigned int)((ga >> 32) & 0x01FFFFFFull) | 0x80000000u; // addr[56:32], type=2
    i32x8 g1;
    g1[0] = 0x07920000;       // data_size=4B, pad_enable, interval=128dw, amount=4dw
    g1[1] = (int)(DMOD << 16);                      // tensor_dim0 = 1024 (lo16)
    g1[2] = (int)(LSEQ << 16);                      // dim0 hi16=0 | tensor_dim1=4096 lo16
    g1[3] = (int)(TGW << 16);                       // dim1 hi16=0 | tile_dim0 = 128
    g1[4] = 16;                                     // tile_dim1 = #row indices
    g1[5] = DMOD;                                   // tensor_dim0_stride = 1024
    g1[6] = 0;
    g1[7] = 0;
    i32x4 g2, g3;
#pragma unroll
    for (int k = 0; k < 4; k++) {
      g2[k] = (tok[2 * k] & 0xFFFF) | (tok[2 * k + 1] << 16);
      g3[k] = (tok[8 + 2 * k] & 0xFFFF) | (tok[8 + 2 * k + 1] << 16);
    }
#if __clang_major__ >= 23
    i32x8 gz = {};
    __builtin_amdgcn_tensor_load_to_lds(g0, g1, g2, g3, gz, 0);
#else
    __builtin_amdgcn_tensor_load_to_lds(g0, g1, g2, g3, 0);
#endif
  }
#endif

  if (tg == 0 && tid < CHLEN) {                     // inclusive decay (exact products)
    float d = 1.0f;
    for (int k = 0; k <= tid; k++) d *= aS[k];
    decrow[(size_t)b * LSEQ + c * CHLEN + tid] = d;
    if (tid == CHLEN - 1) Achunk[b * NCHMAX + c] = d;   // whole-chunk decay
  }

  {                                                 // W[t][s] = p_s * prod_{k=s+1..t} a_k
    const int t = tid >> 4, s = tid & 15;
    float w = 0.0f;
    if (s <= t) { w = pS[s]; for (int k = s + 1; k <= t; k++) w *= aS[k]; }
    W[t][s] = w;
  }

#if USE_TDM
  if (tid < 32) __builtin_amdgcn_s_wait_tensorcnt(0);   // DMA landed (wave 0)
#else
  // fallback: cooperative float4 gather of the 16 rows x 128 channels
  const float* hb = hid + (size_t)b * LSEQ * DMOD + (size_t)tg * TGW;
  for (int i = tid; i < CHLEN * TGW / 4; i += 256) {
    const int row = i >> 5;                         // 32 float4 per row
    const int c4  = i & 31;
    const float4 v = *(const float4*)(hb + (size_t)tokS[row] * DMOD + c4 * 4);
    *(float4*)(&H[row][c4 * 4]) = v;
  }
#endif
  __syncthreads();

  // per-wave 16x16 output tile; A/B layouts per CDNA5 ISA 16x4 f32 striping
  const int wave  = tid >> 5, lane = tid & 31;
  const int mrow  = lane & 15;                      // A: M, B/D: N
  const int khalf = (lane < 16) ? 0 : 2;            // K offset from lane half
  const int ncol  = wave * CHLEN + mrow;            // column within H slab

  v8f acc = {};
#pragma unroll
  for (int kk = 0; kk < 4; kk++) {
    v2f Af, Bf;
    Af.x = W[mrow][kk * 4 + khalf + 0];
    Af.y = W[mrow][kk * 4 + khalf + 1];
    Bf.x = H[kk * 4 + khalf + 0][ncol];
    Bf.y = H[kk * 4 + khalf + 1][ncol];
    acc = __builtin_amdgcn_wmma_f32_16x16x4_f32(false, Af, false, Bf,
                                                (short)0, acc, false, false);
  }

  // store D: VGPR r -> M = r (lanes 0-15) / r+8 (lanes 16-31), N = lane&15
  const int gcol = tg * TGW + ncol;
  float* Yb = Ybuf + ((size_t)b * LSEQ + (size_t)c * CHLEN) * DMOD + gcol;
  const int mbase = (lane < 16) ? 0 : 8;
#pragma unroll
  for (int r = 0; r < 8; r++) Yb[(size_t)(mbase + r) * DMOD] = acc[r];

  if (lane >= 16)                                   // row M=15 = chunk-local end state
    cstate[((size_t)b * NCHMAX + c) * DMOD + gcol] = acc[7];
}

// ---------------------------------------------------------------------------
// Stage 3: sequential inter-chunk carry scan (scalar coeff, vector state).
//   carry[c] = state entering chunk c; S_c = A_c*S_{c-1} + s_c.
//   Deliberately sequential over chunks: global prefix products of (1-p)
//   underflow f32 after ~100 tokens, so no all-prefix-product trick.
// ---------------------------------------------------------------------------
__global__ void __launch_bounds__(256)
k_carry(const float* __restrict__ cstate, const float* __restrict__ Achunk,
        const int* __restrict__ Karr, float* __restrict__ carry) {
  const int b  = blockIdx.y;
  const int ch = blockIdx.x * 256 + threadIdx.x;
  const int K = Karr[b];
  const int nchunk = (K + CHLEN - 1) / CHLEN;
  float s = 0.0f;
  for (int c = 0; c < nchunk; c++) {
    if (c + 1 < nchunk)
      __builtin_prefetch(cstate + ((size_t)b * NCHMAX + c + 1) * DMOD + ch, 0, 1);
    carry[((size_t)b * NCHMAX + c) * DMOD + ch] = s;
    s = fmaf(Achunk[b * NCHMAX + c], s, cstate[((size_t)b * NCHMAX + c) * DMOD + ch]);
  }
}

// ---------------------------------------------------------------------------
// Stage 4: fused finalize + broadcast gather (dominant traffic, ~32 MiB
// coalesced writes; Y rows are L2-resident on a 192 MB L2).
//   out[b,t,:] = Y[j] + decrow[j] * carry[j/16],  j = idxmap[b,t].
// ---------------------------------------------------------------------------
__global__ void __launch_bounds__(256)
k_out(const float* __restrict__ Ybuf, const float* __restrict__ carry,
      const float* __restrict__ decrow, const int* __restrict__ idxmap,
      float* __restrict__ out) {
  const int t = blockIdx.x, b = blockIdx.y;
  const int d4 = threadIdx.x;                       // 256 x float4 = 1024 ch
  const int j = idxmap[(size_t)b * LSEQ + t];
  const int c = j >> 4;
  const float dec = decrow[(size_t)b * LSEQ + j];
  const float4 y = *(const float4*)(Ybuf  + ((size_t)b * LSEQ   + j) * DMOD + d4 * 4);
  const float4 s = *(const float4*)(carry + ((size_t)b * NCHMAX + c) * DMOD + d4 * 4);
  float4 o;
  o.x = fmaf(dec, s.x, y.x);
  o.y = fmaf(dec, s.y, y.y);
  o.z = fmaf(dec, s.z, y.z);
  o.w = fmaf(dec, s.w, y.w);
  *(float4*)(out + ((size_t)b * LSEQ + t) * DMOD + d4 * 4) = o;
}

// ---------------------------------------------------------------------------
extern "C" void kernel_launch(void* const* d_in, const int* in_sizes, int n_in,
                              void* d_out, int out_size, void* d_ws, size_t ws_size,
                              hipStream_t stream) {
  const float*         hid  = (const float*)d_in[0];          // (2,4096,1024) f32
  const float*         prob = (const float*)d_in[1];          // (2,4096) f32
  const unsigned char* mask = (const unsigned char*)d_in[2];  // (2,4096) bool

  char* ws = (char*)d_ws;
  int*   selIdx = (int*)  (ws + SELIDX_OFF);
  float* pcoef  = (float*)(ws + PCOEF_OFF);
  float* decrow = (float*)(ws + DECROW_OFF);
  int*   idxmap = (int*)  (ws + IDXMAP_OFF);
  int*   Karr   = (int*)  (ws + KARR_OFF);
  float* Achunk = (float*)(ws + ACHUNK_OFF);
  float* cstate = (float*)(ws + CSTATE_OFF);
  float* carry  = (float*)(ws + CARRY_OFF);
  float* Ybuf   = (float*)(ws + YBUF_OFF);
  float* out    = (float*)d_out;

  k_setup<<<dim3(BSZ), dim3(1024), 0, stream>>>(prob, mask, selIdx, pcoef, idxmap, Karr);
  k_chunk<<<dim3(TGROUPS, NCHMAX, BSZ), dim3(256), 0, stream>>>(
      hid, selIdx, pcoef, Karr, Ybuf, cstate, Achunk, decrow);
  k_carry<<<dim3(DMOD / 256, BSZ), dim3(256), 0, stream>>>(cstate, Achunk, Karr, carry);
  k_out<<<dim3(LSEQ, BSZ), dim3(256), 0, stream>>>(Ybuf, carry, decrow, idxmap, out);
}